// RecursiveFilter_48395691491666
// MI455X (gfx1250) — compile-verified
//
#include <hip/hip_runtime.h>

// Recursive filter y[n] = W*x[n] + (1-W)*y[n-1] along N axis of [B,C,N,H,W],
// recast as Y = L @ X per (b,c) slab with L a constant 32x32 lower-triangular
// matrix, computed with V_WMMA_F32_16X16X4_F32 (full f32 precision).

typedef __attribute__((ext_vector_type(2))) float v2f;
typedef __attribute__((ext_vector_type(8))) float v8f;

#define FILT_W 0.3f
#define FILT_Q 0.7f   // 1 - W

__device__ __forceinline__ float qpow(int e) {
    float p = 1.0f;
    for (int i = 0; i < e; ++i) p *= FILT_Q;
    return p;
}

// L[i][j]: i = output frame, j = input frame
__device__ __forceinline__ float Lcoef(int i, int j) {
    if (j > i) return 0.0f;
    float p = qpow(i - j);
    return (j == 0) ? p : (FILT_W * p);
}

__global__ __launch_bounds__(256) void
RecursiveFilter_48395691491666_kernel(const float* __restrict__ x,
                                      float* __restrict__ y) {
    constexpr int N   = 32;
    constexpr int S   = 256 * 256;              // spatial columns per slab
    constexpr int TPW = 4;                      // 16-col tiles per wave
    constexpr int jobsPerSlab = (S / 16) / TPW; // 1024

    const int lane  = threadIdx.x & 31;
    const int gwave = blockIdx.x * (blockDim.x >> 5) + (threadIdx.x >> 5);
    const int slab  = gwave / jobsPerSlab;      // (b,c) index, 0..BC-1
    const int job   = gwave % jobsPerSlab;

    const long slabBase = (long)slab * N * S;

    const int m    = lane & 15;   // A row (mod 16) == B/D column (mod 16)
    const int half = lane >> 4;   // lane half selects K sub-pair / D row half
    const int koff = half * 2;

    // ---- A fragments: constant L tiles, computed once per wave ----
    // f32 16x4 A layout: VGPR0 = A[m][k0 + 2*half], VGPR1 = A[m][k0 + 2*half + 1]
    v2f aF[2][8];
#pragma unroll
    for (int mt = 0; mt < 2; ++mt) {
#pragma unroll
        for (int ks = 0; ks < 8; ++ks) {
            const int row = mt * 16 + m;
            const int k   = ks * 4 + koff;
            aF[mt][ks].x = Lcoef(row, k + 0);
            aF[mt][ks].y = Lcoef(row, k + 1);
        }
    }

    const int tile0 = job * TPW;
    for (int t = 0; t < TPW; ++t) {
        const int col = (tile0 + t) * 16 + m;
        const float* __restrict__ px = x + slabBase + col;
        float*       __restrict__ py = y + slabBase + col;

        v8f acc0 = {};  // output frames 0..15
        v8f acc1 = {};  // output frames 16..31

#pragma unroll
        for (int ks = 0; ks < 8; ++ks) {
            // B 4x16 fragment: VGPR0 = X[k0 + 2*half][col], VGPR1 = X[k0 + 2*half + 1][col]
            const int r = ks * 4 + koff;
            v2f bF;
            bF.x = px[(long)(r + 0) * S];
            bF.y = px[(long)(r + 1) * S];

            acc0 = __builtin_amdgcn_wmma_f32_16x16x4_f32(
                false, aF[0][ks], false, bF, (short)0, acc0, false, false);
            acc1 = __builtin_amdgcn_wmma_f32_16x16x4_f32(
                false, aF[1][ks], false, bF, (short)0, acc1, false, false);
        }

        // D layout: VGPR v holds row (v + 8*half), column col
#pragma unroll
        for (int v = 0; v < 8; ++v) {
            const int r = v + half * 8;
            py[(long)r * S]        = acc0[v];
            py[(long)(16 + r) * S] = acc1[v];
        }
    }
}

extern "C" void kernel_launch(void* const* d_in, const int* in_sizes, int n_in,
                              void* d_out, int out_size, void* d_ws, size_t ws_size,
                              hipStream_t stream) {
    (void)n_in; (void)out_size; (void)d_ws; (void)ws_size;
    const float* x = (const float*)d_in[0];
    float*       y = (float*)d_out;

    constexpr int N = 32;
    constexpr int S = 256 * 256;
    const int BC = in_sizes[0] / (N * S);          // 12 for [4,3,32,256,256]

    constexpr int jobsPerSlab  = (S / 16) / 4;     // 1024 wave-jobs per slab
    constexpr int wavesPerBlk  = 8;                // 256 threads, wave32
    const int totalWaves = BC * jobsPerSlab;       // 12288
    const int blocks     = totalWaves / wavesPerBlk;

    RecursiveFilter_48395691491666_kernel<<<blocks, 256, 0, stream>>>(x, y);
}